// BoundaryLoss_45518063403454
// MI455X (gfx1250) — compile-verified
//
#include <hip/hip_runtime.h>

// BoundaryLoss: out = mean_i relu(0.99999 - inputs[i, labels[i]])
// N = 65536 rows, G = 1024 cols, labels int64 (only low dword needed).
// Gather-latency-bound: ~65536 scattered 4B loads (one cacheline each).
// WMMA f32 16x16x4 with ones-B is used as the wave-horizontal reduction engine.

typedef __attribute__((ext_vector_type(2))) float v2f;
typedef __attribute__((ext_vector_type(8))) float v8f;

#define POS_MARGIN 0.99999f

// Full-wave (32-lane) sum via two V_WMMA_F32_16X16X4_F32 with all-ones B.
// Requires EXEC == all ones (call from converged full waves only).
// Step 1: A[m][0]=s(lane m), A[m][2]=s(lane m+16) -> D[m][*] = s[m]+s[m+16] = t[m]
//   D layout: lanes 0-15 see d[v]=t[v]; lanes 16-31 see d[v]=t[8+v]
// Sum 8 components -> low lanes hold sum t[0..7], high lanes sum t[8..15]
// Step 2: same trick folds the halves -> full wave sum in every lane.
__device__ __forceinline__ float wave_sum_wmma(float s) {
    v2f b; b.x = 1.0f; b.y = 1.0f;   // all-ones B (layout-independent)
    v8f c = {};                      // zero accumulator

    v2f a; a.x = s; a.y = 0.0f;
    v8f d = __builtin_amdgcn_wmma_f32_16x16x4_f32(
        /*neg_a=*/false, a, /*neg_b=*/false, b,
        /*c_mod=*/(short)0, c, /*reuse_a=*/false, /*reuse_b=*/false);

    float u = d[0] + d[1] + d[2] + d[3] + d[4] + d[5] + d[6] + d[7];

    v2f a2; a2.x = u; a2.y = 0.0f;
    v8f d2 = __builtin_amdgcn_wmma_f32_16x16x4_f32(
        false, a2, false, b, (short)0, c, false, false);

    return d2[0];  // full-wave sum, replicated in every lane
}

#define BLK 256
#define ROWS_PER_THREAD 4

// Pass 1: gather + per-block partial sum -> partials[blockIdx.x]
// labels_lo: int64 labels viewed as int32 pairs; low dword at index 2*row.
__global__ void __launch_bounds__(BLK)
boundary_loss_partial(const float* __restrict__ inputs,
                      const int* __restrict__ labels_lo,
                      float* __restrict__ partials,
                      int N, int G) {
    const int tid  = threadIdx.x;
    const int base = blockIdx.x * (BLK * ROWS_PER_THREAD) + tid;

    int   row[ROWS_PER_THREAD];
    int   lab[ROWS_PER_THREAD];
    float val[ROWS_PER_THREAD];

    // Batch 1: all label loads in flight (coalesced 4B, stride 8B per lane)
#pragma unroll
    for (int k = 0; k < ROWS_PER_THREAD; ++k) {
        const int r = base + k * BLK;
        row[k] = (r < N) ? r : (N - 1);          // clamp: loads stay in-bounds
        lab[k] = labels_lo[2 * row[k]];
    }

    // Batch 2: all gathers in flight (32-bit index: row*G+lab < 2^26)
#pragma unroll
    for (int k = 0; k < ROWS_PER_THREAD; ++k)
        val[k] = inputs[(unsigned)(row[k] * G) + (unsigned)lab[k]];

    // Batch 3: accumulate; tail rows contribute 0 without diverging EXEC
    float s = 0.0f;
#pragma unroll
    for (int k = 0; k < ROWS_PER_THREAD; ++k) {
        const float contrib = fmaxf(POS_MARGIN - val[k], 0.0f);
        s += ((base + k * BLK) < N) ? contrib : 0.0f;
    }

    // converged, EXEC all ones: WMMA wave reduction
    const float wsum = wave_sum_wmma(s);

    __shared__ float lds[BLK / 32];
    const int wave = tid >> 5;
    const int lane = tid & 31;
    if (lane == 0) lds[wave] = wsum;
    __syncthreads();

    if (tid == 0) {
        float bs = 0.0f;
#pragma unroll
        for (int w = 0; w < BLK / 32; ++w) bs += lds[w];  // fixed order: deterministic
        partials[blockIdx.x] = bs;
    }
}

// Pass 2: single wave folds all block partials, scales by 1/N.
__global__ void __launch_bounds__(32)
boundary_loss_final(const float* __restrict__ partials,
                    float* __restrict__ out,
                    int nblocks, int N) {
    const int lane = threadIdx.x;  // one wave of 32
    float s = 0.0f;
    for (int i = lane; i < nblocks; i += 32)   // fixed order per lane: deterministic
        s += partials[i];

    const float tot = wave_sum_wmma(s);        // full wave, EXEC all ones
    if (lane == 0)
        out[0] = tot * (1.0f / (float)N);
}

extern "C" void kernel_launch(void* const* d_in, const int* in_sizes, int n_in,
                              void* d_out, int out_size, void* d_ws, size_t ws_size,
                              hipStream_t stream) {
    const float* inputs    = (const float*)d_in[0];
    const int*   labels_lo = (const int*)d_in[1];   // int64 viewed as dword pairs
    float* out      = (float*)d_out;
    float* partials = (float*)d_ws;

    const int N = in_sizes[1];                 // 65536
    const int G = in_sizes[0] / N;             // 1024

    const int rows_per_block = BLK * ROWS_PER_THREAD;
    const int nblocks = (N + rows_per_block - 1) / rows_per_block;  // 64

    boundary_loss_partial<<<nblocks, BLK, 0, stream>>>(inputs, labels_lo, partials, N, G);
    boundary_loss_final<<<1, 32, 0, stream>>>(partials, out, nblocks, N);
}